// KMeansLayer_13374528160286
// MI455X (gfx1250) — compile-verified
//
#include <hip/hip_runtime.h>

// KMeans (Lloyd) per frame: B=32, N=256*256 pixels, C=32 channels, K=8, 10 iters.
// One workgroup per frame (frame x = 8MB -> L2 resident across all passes).
// Distance GEMM: D = Cent(16x32, 8 zero rows) @ X_tile^T(32x16) via exact-fp32
// V_WMMA_F32_16X16X4_F32, so each lane 0..15 holds all 8 cluster scores for its
// own pixel -> per-lane branchless argmin, no cross-lane butterfly.

#define K_CL   8
#define C_CH   32
#define N_ITER 10
#define NPIX   (256 * 256)
#define TILES  (NPIX / 16)      // 4096 tiles of 16 pixels
#define THREADS 512
#define WAVES   (THREADS / 32)

typedef float v2f __attribute__((ext_vector_type(2)));
typedef float v8f __attribute__((ext_vector_type(8)));

__global__ __launch_bounds__(THREADS)
void KMeansLayer_kernel(const float* __restrict__ x,
                        const int* __restrict__ init_idx,
                        float* __restrict__ out)
{
    __shared__ float cent[K_CL][C_CH];   // centroids
    __shared__ float sums[K_CL][C_CH];   // per-cluster channel sums
    __shared__ float counts[K_CL];       // per-cluster pixel counts (float)
    __shared__ float c2s[K_CL];          // |cent_k|^2
    __shared__ float means[K_CL];        // final scalar means

    const int b    = blockIdx.x;
    const float* xb = x + (size_t)b * NPIX * C_CH;
    const int tid  = threadIdx.x;
    const int lane = tid & 31;
    const int wave = tid >> 5;
    const int pix  = lane & 15;          // pixel column owned by this lane (D layout)
    const int hi   = (lane >> 4) & 1;    // K-pair select within A/B fragments

    // ---- init centroids from init_idx ----
    if (tid < K_CL * C_CH) {
        int k = tid / C_CH, c = tid % C_CH;
        int p = init_idx[b * K_CL + k];
        cent[k][c] = xb[(size_t)p * C_CH + c];
    }
    __syncthreads();

    // N_ITER update passes + 1 final accumulation pass (it == N_ITER)
    for (int it = 0; it <= N_ITER; ++it) {
        if (tid < K_CL) {
            float s = 0.f;
            #pragma unroll
            for (int c = 0; c < C_CH; ++c) { float v = cent[tid][c]; s += v * v; }
            c2s[tid]    = s;
            counts[tid] = 0.f;
        }
        if (tid < K_CL * C_CH) sums[tid / C_CH][tid % C_CH] = 0.f;
        __syncthreads();

        // A fragments: M = cluster (rows 8..15 zero), loop-invariant this pass.
        v2f af[8];
        #pragma unroll
        for (int s = 0; s < 8; ++s) {
            int k0 = 4 * s + 2 * hi;
            float a0 = (pix < K_CL) ? cent[pix][k0]     : 0.f;
            float a1 = (pix < K_CL) ? cent[pix][k0 + 1] : 0.f;
            af[s] = (v2f){a0, a1};
        }
        float c2r[K_CL];
        #pragma unroll
        for (int r = 0; r < K_CL; ++r) c2r[r] = c2s[r];

        for (int t = wave; t < TILES; t += WAVES) {
            const float* xr = xb + (size_t)(t * 16 + pix) * C_CH;
            if (t + WAVES < TILES)
                __builtin_prefetch(xb + (size_t)(t + WAVES) * 16 * C_CH + pix * C_CH, 0, 1);

            // B fragments: N = pixel, K = channel pairs.
            v2f bx[8];
            #pragma unroll
            for (int s = 0; s < 8; ++s)
                bx[s] = *(const v2f*)(xr + 4 * s + 2 * hi);

            // acc[r] (lanes 0..15) = dot(cent[r], X[pixel = lane])
            v8f acc = {};
            #pragma unroll
            for (int s = 0; s < 8; ++s)
                acc = __builtin_amdgcn_wmma_f32_16x16x4_f32(
                        false, af[s], false, bx[s], (short)0, acc, false, false);

            // Per-lane branchless argmin over 8 clusters (first-min tie-break).
            float best = __builtin_fmaf(-2.f, acc[0], c2r[0]);
            int bi = 0;
            #pragma unroll
            for (int r = 1; r < K_CL; ++r) {
                float sc = __builtin_fmaf(-2.f, acc[r], c2r[r]);
                bool better = sc < best;
                bi   = better ? r  : bi;
                best = better ? sc : best;
            }
            // Broadcast pixel's assignment from lane (L&15) to lane L (one DS op).
            int a32 = __builtin_amdgcn_ds_swizzle(bi, 0x000F);

            // Segment sums straight from B fragments already in registers.
            #pragma unroll
            for (int s = 0; s < 8; ++s) {
                int k0 = 4 * s + 2 * hi;
                atomicAdd(&sums[a32][k0],     bx[s].x);   // ds_add_f32
                atomicAdd(&sums[a32][k0 + 1], bx[s].y);
            }
            if (lane < 16) atomicAdd(&counts[a32], 1.0f);
        }
        __syncthreads();

        if (it < N_ITER) {
            // centroid update; keep empty clusters
            if (tid < K_CL * C_CH) {
                int k = tid / C_CH, c = tid % C_CH;
                float cnt = counts[k];
                cent[k][c] = (cnt > 0.f) ? (sums[k][c] / cnt) : cent[k][c];
            }
        } else {
            // final scalar means: sum over channels / (count * C)
            if (tid < K_CL) {
                float s = 0.f;
                #pragma unroll
                for (int c = 0; c < C_CH; ++c) s += sums[tid][c];
                means[tid] = s / fmaxf(counts[tid] * (float)C_CH, 1.0f);
            }
        }
        __syncthreads();
    }

    // ---- output pass: recompute final assignment, write means[assign] ----
    {
        v2f af[8];
        #pragma unroll
        for (int s = 0; s < 8; ++s) {
            int k0 = 4 * s + 2 * hi;
            float a0 = (pix < K_CL) ? cent[pix][k0]     : 0.f;
            float a1 = (pix < K_CL) ? cent[pix][k0 + 1] : 0.f;
            af[s] = (v2f){a0, a1};
        }
        float c2r[K_CL];
        #pragma unroll
        for (int r = 0; r < K_CL; ++r) c2r[r] = c2s[r];
        float* outb = out + (size_t)b * NPIX;

        for (int t = wave; t < TILES; t += WAVES) {
            const float* xr = xb + (size_t)(t * 16 + pix) * C_CH;
            v2f bx[8];
            #pragma unroll
            for (int s = 0; s < 8; ++s)
                bx[s] = *(const v2f*)(xr + 4 * s + 2 * hi);

            v8f acc = {};
            #pragma unroll
            for (int s = 0; s < 8; ++s)
                acc = __builtin_amdgcn_wmma_f32_16x16x4_f32(
                        false, af[s], false, bx[s], (short)0, acc, false, false);

            float best = __builtin_fmaf(-2.f, acc[0], c2r[0]);
            int bi = 0;
            #pragma unroll
            for (int r = 1; r < K_CL; ++r) {
                float sc = __builtin_fmaf(-2.f, acc[r], c2r[r]);
                bool better = sc < best;
                bi   = better ? r  : bi;
                best = better ? sc : best;
            }
            if (lane < 16)
                outb[t * 16 + lane] = means[bi];   // lane owns its pixel's assign
        }
    }
}

extern "C" void kernel_launch(void* const* d_in, const int* in_sizes, int n_in,
                              void* d_out, int out_size, void* d_ws, size_t ws_size,
                              hipStream_t stream) {
    const float* x        = (const float*)d_in[0];
    const int*   init_idx = (const int*)d_in[1];
    float*       out      = (float*)d_out;
    const int B = in_sizes[1] / K_CL;   // 32 frames
    KMeansLayer_kernel<<<dim3(B), dim3(THREADS), 0, stream>>>(x, init_idx, out);
}